// AdaptiveGNN_76175539962271
// MI455X (gfx1250) — compile-verified
//
#include <hip/hip_runtime.h>
#include <math.h>

typedef __attribute__((ext_vector_type(2))) float v2f;
typedef __attribute__((ext_vector_type(8))) float v8f;

#define LN_EPS 1e-5f

__device__ __forceinline__ void atomic_add_f32(float* p, float v) {
    __hip_atomic_fetch_add(p, v, __ATOMIC_RELAXED, __HIP_MEMORY_SCOPE_AGENT);
}

// ---------------- degree / norm ----------------
__global__ void k_fill_ones(float* __restrict__ deg, int n) {
    int i = blockIdx.x * blockDim.x + threadIdx.x;
    if (i < n) deg[i] = 1.0f;  // self-loop contribution
}

__global__ void k_degree(const int* __restrict__ dst, float* __restrict__ deg, int E) {
    int e = blockIdx.x * blockDim.x + threadIdx.x;
    if (e < E) atomic_add_f32(&deg[dst[e]], 1.0f);
}

__global__ void k_rsqrt_inplace(float* __restrict__ d, int n) {
    int i = blockIdx.x * blockDim.x + threadIdx.x;
    if (i < n) d[i] = rsqrtf(d[i]);
}

// ---------------- fp32 WMMA GEMM: H[N,F] = X[N,K] @ W[K,F] ----------------
// One wave computes a 16x64 output tile (4x v8f accumulators), K-step = 4.
// Requires N%16==0, F%64==0, K%4==0 (true here: N=100000, K in {128,256}, F in {256,128}).
__global__ __launch_bounds__(32)
void k_gemm_wmma_f32(const float* __restrict__ X, const float* __restrict__ W,
                     float* __restrict__ H, int K, int F) {
    const int lane = threadIdx.x;       // 0..31
    const int m    = lane & 15;
    const int kh   = lane >> 4;         // 0: K=k0,k0+1 ; 1: K=k0+2,k0+3
    const int row0 = blockIdx.y << 4;   // 16 rows of output
    const int col0 = blockIdx.x << 6;   // 64 cols of output

    const float* xrow = X + (size_t)(row0 + m) * K + 2 * kh;

    v8f acc0 = {}, acc1 = {}, acc2 = {}, acc3 = {};

    for (int k0 = 0; k0 < K; k0 += 4) {
        // A fragment (16x4), reused for all 4 column tiles
        v2f a;
        a.x = xrow[k0];
        a.y = xrow[k0 + 1];

        // B fragments (4x16 each): rows (k0+2*kh, k0+2*kh+1) per lane half
        const float* wr = W + (size_t)(k0 + 2 * kh) * F + col0 + m;
        v2f b0; b0.x = wr[0];  b0.y = wr[F + 0];
        v2f b1; b1.x = wr[16]; b1.y = wr[F + 16];
        v2f b2; b2.x = wr[32]; b2.y = wr[F + 32];
        v2f b3; b3.x = wr[48]; b3.y = wr[F + 48];

        acc0 = __builtin_amdgcn_wmma_f32_16x16x4_f32(false, a, false, b0, (short)0, acc0, false, false);
        acc1 = __builtin_amdgcn_wmma_f32_16x16x4_f32(false, a, false, b1, (short)0, acc1, false, false);
        acc2 = __builtin_amdgcn_wmma_f32_16x16x4_f32(false, a, false, b2, (short)0, acc2, false, false);
        acc3 = __builtin_amdgcn_wmma_f32_16x16x4_f32(false, a, false, b3, (short)0, acc3, false, false);
    }

    // C/D layout: lanes 0-15 -> rows row0+v, lanes 16-31 -> rows row0+8+v (v = VGPR index)
    float* hp = H + (size_t)(row0 + kh * 8) * F + col0 + m;
#pragma unroll
    for (int v = 0; v < 8; ++v) {
        hp[(size_t)v * F + 0]  = acc0[v];
        hp[(size_t)v * F + 16] = acc1[v];
        hp[(size_t)v * F + 32] = acc2[v];
        hp[(size_t)v * F + 48] = acc3[v];
    }
}

// ---------------- out[i,:] = bias + h[i,:] * dinv[i]^2  (self-loop + bias init) ----------------
__global__ void k_bias_selfloop(const float* __restrict__ h, const float* __restrict__ bias,
                                const float* __restrict__ dinv, float* __restrict__ out,
                                int n, int F, int fshift) {
    long long t = (long long)blockIdx.x * blockDim.x + threadIdx.x;
    long long total = (long long)n << fshift;   // n * (F/4)
    if (t >= total) return;
    int i  = (int)(t >> fshift);
    int fc = (int)(t & ((1 << fshift) - 1));
    float di = dinv[i];
    float w  = di * di;
    const float4 hv = *(const float4*)(h + (size_t)i * F + fc * 4);
    const float4 bv = *(const float4*)(bias + fc * 4);
    float4 o;
    o.x = bv.x + hv.x * w;
    o.y = bv.y + hv.y * w;
    o.z = bv.z + hv.z * w;
    o.w = bv.w + hv.w * w;
    *(float4*)(out + (size_t)i * F + fc * 4) = o;
}

// ---------------- edge scatter: out[dst,:] += h[src,:] * dinv[src]*dinv[dst] ----------------
__global__ void k_edge_scatter(const float* __restrict__ h, const int* __restrict__ src,
                               const int* __restrict__ dst, const float* __restrict__ dinv,
                               float* __restrict__ out, int E, int F, int fshift) {
    long long t = (long long)blockIdx.x * blockDim.x + threadIdx.x;
    int e = (int)(t >> fshift);
    if (e >= E) return;
    int fc = (int)(t & ((1 << fshift) - 1));
    int s = src[e];
    int d = dst[e];
    float nrm = dinv[s] * dinv[d];
    const float4 hv = *(const float4*)(h + (size_t)s * F + fc * 4);
    float* op = out + (size_t)d * F + fc * 4;
    atomic_add_f32(op + 0, hv.x * nrm);
    atomic_add_f32(op + 1, hv.y * nrm);
    atomic_add_f32(op + 2, hv.z * nrm);
    atomic_add_f32(op + 3, hv.w * nrm);
}

// ---------------- x = elu( LN(x * sigmoid(a)) * g + beta ), F = 256, one wave per node ----------------
__global__ __launch_bounds__(256)
void k_gate_ln_elu(float* __restrict__ x, const float* __restrict__ aptr,
                   const float* __restrict__ g, const float* __restrict__ beta, int n) {
    int node = blockIdx.x * 8 + (threadIdx.x >> 5);
    int lane = threadIdx.x & 31;
    if (node >= n) return;  // uniform per wave
    float gate = 1.0f / (1.0f + __expf(-aptr[0]));
    float* xp = x + (size_t)node * 256;

    float v[8];
    float s = 0.f, sq = 0.f;
#pragma unroll
    for (int j = 0; j < 8; ++j) {
        float t = xp[lane + 32 * j] * gate;
        v[j] = t;
        s += t;
        sq += t * t;
    }
#pragma unroll
    for (int msk = 16; msk > 0; msk >>= 1) {
        s  += __shfl_xor(s, msk, 32);
        sq += __shfl_xor(sq, msk, 32);
    }
    float mu  = s * (1.0f / 256.0f);
    float var = sq * (1.0f / 256.0f) - mu * mu;
    float rs  = rsqrtf(var + LN_EPS);
#pragma unroll
    for (int j = 0; j < 8; ++j) {
        int f = lane + 32 * j;
        float t = (v[j] - mu) * rs * g[f] + beta[f];
        xp[f] = (t > 0.f) ? t : (__expf(t) - 1.f);
    }
}

extern "C" void kernel_launch(void* const* d_in, const int* in_sizes, int n_in,
                              void* d_out, int out_size, void* d_ws, size_t ws_size,
                              hipStream_t stream) {
    const float* x   = (const float*)d_in[0];
    const int*   ei  = (const int*)d_in[1];
    const float* W0  = (const float*)d_in[2];
    const float* b0  = (const float*)d_in[3];
    const float* W1  = (const float*)d_in[4];
    const float* b1  = (const float*)d_in[5];
    const float* W2  = (const float*)d_in[6];
    const float* b2  = (const float*)d_in[7];
    const float* W3  = (const float*)d_in[8];
    const float* b3  = (const float*)d_in[9];
    const float* a0  = (const float*)d_in[10];
    const float* a1  = (const float*)d_in[11];
    const float* a2  = (const float*)d_in[12];
    const float* g0  = (const float*)d_in[13];
    const float* be0 = (const float*)d_in[14];
    const float* g1  = (const float*)d_in[15];
    const float* be1 = (const float*)d_in[16];
    const float* g2  = (const float*)d_in[17];
    const float* be2 = (const float*)d_in[18];

    const int N = in_sizes[0] / 128;
    const int E = in_sizes[1] / 2;
    const int* src = ei;
    const int* dst = ei + E;

    // workspace layout: dinv[N] | H[N*256] | AGG[N*256]
    float* ws   = (float*)d_ws;
    float* dinv = ws;
    size_t off  = (((size_t)N + 63) & ~(size_t)63);
    float* H    = ws + off;
    float* AGG  = H + (size_t)N * 256;
    float* out  = (float*)d_out;

    // ---- symmetric GCN normalization ----
    k_fill_ones<<<(N + 255) / 256, 256, 0, stream>>>(dinv, N);
    k_degree<<<(E + 255) / 256, 256, 0, stream>>>(dst, dinv, E);
    k_rsqrt_inplace<<<(N + 255) / 256, 256, 0, stream>>>(dinv, N);

    // ---- one GCN conv layer: obuf = scatter(in @ W) + bias ----
    auto run_layer = [&](const float* in, int K, const float* W, const float* b,
                         int F, float* hbuf, float* obuf) {
        dim3 grid(F / 64, N / 16);
        k_gemm_wmma_f32<<<grid, 32, 0, stream>>>(in, W, hbuf, K, F);
        int fshift = (F == 256) ? 6 : 5;            // log2(F/4)
        long long tinit = (long long)N << fshift;
        k_bias_selfloop<<<(unsigned)((tinit + 255) / 256), 256, 0, stream>>>(
            hbuf, b, dinv, obuf, N, F, fshift);
        long long tedge = (long long)E << fshift;
        k_edge_scatter<<<(unsigned)((tedge + 255) / 256), 256, 0, stream>>>(
            hbuf, src, dst, dinv, obuf, E, F, fshift);
    };

    // layer 0: 128 -> 256
    run_layer(x, 128, W0, b0, 256, H, AGG);
    k_gate_ln_elu<<<(N + 7) / 8, 256, 0, stream>>>(AGG, a0, g0, be0, N);
    // layer 1: 256 -> 256
    run_layer(AGG, 256, W1, b1, 256, H, AGG);
    k_gate_ln_elu<<<(N + 7) / 8, 256, 0, stream>>>(AGG, a1, g1, be1, N);
    // layer 2: 256 -> 256
    run_layer(AGG, 256, W2, b2, 256, H, AGG);
    k_gate_ln_elu<<<(N + 7) / 8, 256, 0, stream>>>(AGG, a2, g2, be2, N);
    // layer 3: 256 -> 128 (no gate/LN/ELU), write straight to d_out
    run_layer(AGG, 256, W3, b3, 128, H, out);
}